// CRF_2405181685864
// MI455X (gfx1250) — compile-verified
//
#include <hip/hip_runtime.h>

// Viterbi CRF decode, fused single kernel for gfx1250 (MI455X).
//
// Max-plus recurrence is not WMMA-expressible (tropical semiring + argmax),
// so the hot loop is VALU with transition columns held in VGPRs. CDNA5-specific
// paths used: global_load_async_to_lds_b128 double-buffered potentials prefetch
// (ASYNCcnt) + s_wait_asynccnt, LDS-resident backpointers (320KB WGP LDS),
// wave32 blocks of 4 waves.

typedef float v4f __attribute__((ext_vector_type(4)));

constexpr int NB = 256;   // batch
constexpr int NT = 1024;  // time
constexpr int NC = 128;   // tags

__global__ __launch_bounds__(128)
void viterbi_fused(const float* __restrict__ x,
                   const float* __restrict__ trans,
                   const int* __restrict__ seq_lens,
                   float* __restrict__ out)
{
    __shared__ __align__(16) float alpha_lds[2][NC];   // double-buffered alpha
    __shared__ __align__(16) float pot_lds[2][NC];     // async-prefetched x[b,t,:]
    __shared__ unsigned char bp_lds[NT][NC];           // backpointers, rows 1..T-1
    __shared__ unsigned char tags_lds[NT];             // decoded tag per t

    const int j = threadIdx.x;           // tag column owned by this thread
    const int b = blockIdx.x;
    const int seq_len = seq_lens[b];
    const float* __restrict__ xb = x + (size_t)b * NT * NC;

    // Transition column j in registers: tr[i] = trans[i][j] (coalesced loads).
    float tr[NC];
#pragma unroll
    for (int i = 0; i < NC; ++i) tr[i] = trans[i * NC + j];

    // alpha_0 = x[b,0,:]
    alpha_lds[0][j] = xb[j];

    // Async prefetch of pot for t=1 into pot_lds[1] (wave 0, 32 lanes x 16B).
    if (j < 32 && seq_len > 1) {
        unsigned lds_a = (unsigned)(size_t)&pot_lds[1][j * 4];
        unsigned voff  = (unsigned)(j * 16 + NC * 4);   // row t=1, chunk j
        asm volatile("global_load_async_to_lds_b128 %0, %1, %2"
                     :: "v"(lds_a), "v"(voff), "s"(xb) : "memory");
    }

    for (int t = 1; t < seq_len; ++t) {
        // pot[t] was issued one iteration ago; wave0 waits, barrier publishes
        // it (and alpha from the previous step) to all 4 waves.
        if (j < 32) asm volatile("s_wait_asynccnt 0" ::: "memory");
        __syncthreads();

        // Prefetch pot[t+1] into the other buffer while we compute step t.
        if (j < 32 && (t + 1) < seq_len) {
            unsigned lds_a = (unsigned)(size_t)&pot_lds[(t + 1) & 1][j * 4];
            unsigned voff  = (unsigned)(j * 16);
            const float* gb = xb + (size_t)(t + 1) * NC;
            asm volatile("global_load_async_to_lds_b128 %0, %1, %2"
                         :: "v"(lds_a), "v"(voff), "s"(gb) : "memory");
        }

        // new_alpha[j] = max_i(alpha[i] + tr[i]) with first-max argmax.
        // 4 independent chains (32-deep each) to expose ILP.
        const v4f* a4 = (const v4f*)&alpha_lds[(t - 1) & 1][0];
        float b0 = -__builtin_inff(), b1 = b0, b2 = b0, b3 = b0;
        int i0 = 0, i1 = 1, i2 = 2, i3 = 3;
#pragma unroll
        for (int q = 0; q < 32; ++q) {
            v4f a = a4[q];   // broadcast ds_load_b128
            float s0 = a.x + tr[4*q+0]; if (s0 > b0) { b0 = s0; i0 = 4*q+0; }
            float s1 = a.y + tr[4*q+1]; if (s1 > b1) { b1 = s1; i1 = 4*q+1; }
            float s2 = a.z + tr[4*q+2]; if (s2 > b2) { b2 = s2; i2 = 4*q+2; }
            float s3 = a.w + tr[4*q+3]; if (s3 > b3) { b3 = s3; i3 = 4*q+3; }
        }
        // Merge chains, preferring the lowest index on exact ties (jnp.argmax).
        float best = b0; int bi = i0;
        if (b1 > best || (b1 == best && i1 < bi)) { best = b1; bi = i1; }
        if (b2 > best || (b2 == best && i2 < bi)) { best = b2; bi = i2; }
        if (b3 > best || (b3 == best && i3 < bi)) { best = b3; bi = i3; }

        alpha_lds[t & 1][j] = best + pot_lds[t & 1][j];
        bp_lds[t][j] = (unsigned char)bi;
    }
    __syncthreads();

    // Final argmax of alpha (redundant on all threads; 128 LDS broadcasts).
    const int fin = (seq_len - 1) & 1;
    float bestA = alpha_lds[fin][0];
    int last_tag = 0;
    for (int i = 1; i < NC; ++i) {
        float v = alpha_lds[fin][i];
        if (v > bestA) { bestA = v; last_tag = i; }
    }

    // For t >= seq_len-1 the backpointers are identity => tag = last_tag.
    for (int tt = seq_len - 1 + j; tt < NT; tt += 128)
        tags_lds[tt] = (unsigned char)last_tag;

    // Serial backtrace at LDS latency (thread 0).
    if (j == 0) {
        int tag = last_tag;
        for (int t = seq_len - 2; t >= 0; --t) {
            tag = bp_lds[t + 1][tag];
            tags_lds[t] = (unsigned char)tag;
        }
    }
    __syncthreads();

    // One-hot expansion: streaming nontemporal b128 stores.
    v4f* out4 = (v4f*)(out + (size_t)b * NT * NC);
#pragma unroll 4
    for (int f4 = j; f4 < NT * NC / 4; f4 += 128) {
        int t  = f4 >> 5;            // 32 float4 per row of 128
        int cb = (f4 & 31) << 2;
        int d  = (int)tags_lds[t] - cb;
        v4f v;
        v.x = (d == 0) ? 1.0f : 0.0f;
        v.y = (d == 1) ? 1.0f : 0.0f;
        v.z = (d == 2) ? 1.0f : 0.0f;
        v.w = (d == 3) ? 1.0f : 0.0f;
        __builtin_nontemporal_store(v, &out4[f4]);
    }
}

extern "C" void kernel_launch(void* const* d_in, const int* in_sizes, int n_in,
                              void* d_out, int out_size, void* d_ws, size_t ws_size,
                              hipStream_t stream) {
    (void)in_sizes; (void)n_in; (void)out_size; (void)d_ws; (void)ws_size;
    const float* x  = (const float*)d_in[0];
    const float* tr = (const float*)d_in[1];
    const int*   sl = (const int*)d_in[2];
    float* out = (float*)d_out;
    viterbi_fused<<<dim3(NB), dim3(128), 0, stream>>>(x, tr, sl, out);
}